// BinaryMasking_89111981457652
// MI455X (gfx1250) — compile-verified
//
#include <hip/hip_runtime.h>
#include <stdint.h>

// Fixed problem configuration (from the reference)
#define TT   16          // T
#define NN   65536       // T*H*W tokens per row
#define BB   128         // batch
#define EPSF 1e-3f
#define HALF_PI 1.5707963267948966f

typedef unsigned int v4u __attribute__((ext_vector_type(4)));
typedef int          v8i __attribute__((ext_vector_type(8)));
typedef int          v4i __attribute__((ext_vector_type(4)));

__device__ __forceinline__ float clampf01(float u) {
  return fminf(fmaxf(u, EPSF), 1.0f - EPSF);
}

// --------------------------------------------------------------------------
// Kernel 1: per-row prep (tiny). Sort the 16 event times, build log tables,
// derive K_src/K_tgt (truncating cast == .astype(int32)) and the dR scalar.
// --------------------------------------------------------------------------
__global__ void prep_kernel(const float* __restrict__ U_event_t,  // [B,T]
                            const float* __restrict__ U_rate,     // [2,B]
                            float* __restrict__ logU,             // [B,T]
                            float* __restrict__ log1mU,           // [B,T]
                            float* __restrict__ dRb,              // [B]
                            int* __restrict__ Ks) {               // [2,B]
  int b = threadIdx.x;
  if (b >= BB) return;
  float v[TT];
#pragma unroll
  for (int t = 0; t < TT; ++t) v[t] = clampf01(U_event_t[b * TT + t]);
  for (int i = 1; i < TT; ++i) {            // insertion sort ascending
    float key = v[i];
    int j = i - 1;
    while (j >= 0 && v[j] > key) { v[j + 1] = v[j]; --j; }
    v[j + 1] = key;
  }
#pragma unroll
  for (int t = 0; t < TT; ++t) {
    logU[b * TT + t]   = logf(v[t]);
    log1mU[b * TT + t] = logf(1.0f - v[t]);
  }
  float u0 = clampf01(U_rate[b]);           // src rate: cosine schedule
  float u1 = clampf01(U_rate[BB + b]);      // tgt rate: linear schedule
  Ks[b]      = (int)((1.0f - cosf(HALF_PI * u0)) * (float)NN);  // trunc toward 0
  Ks[BB + b] = (int)(u1 * (float)NN);
  dRb[b]     = sinf(HALF_PI * u0) * HALF_PI;
}

// --------------------------------------------------------------------------
// Kernel 2: one block per (which, b). TDM-DMA the 256 KB row into the 320 KB
// LDS (with hardware padding: 1 dword per 256 -> bank-conflict-free chunked
// reads), convert to order-preserving u32 keys, exact 4-pass radix-select for
// the K-th largest, stable tie-break, emit the {0,1} mask with float4 stores.
// --------------------------------------------------------------------------
__global__ __launch_bounds__(256)
void topk_mask_kernel(const float* __restrict__ U_base,   // [2,B,N]
                      const float* __restrict__ logU,
                      const float* __restrict__ log1mU,
                      const float* __restrict__ dRb,
                      const int* __restrict__ Ks,
                      float* __restrict__ out) {          // [src|tgt|dR]
  extern __shared__ uint32_t keys[];   // (NN + NN/256) dwords, dynamic LDS
  __shared__ uint32_t hist[256];
  __shared__ uint32_t eqcnt[256];
  __shared__ float    logT[TT];
  __shared__ uint32_t sDigit, sRem;
  __shared__ int      sK;

  const int tid   = threadIdx.x;
  const int which = blockIdx.x >> 7;       // 0 = src, 1 = tgt
  const int b     = blockIdx.x & (BB - 1);

  if (tid < TT) logT[tid] = which ? log1mU[b * TT + tid] : logU[b * TT + tid];
  if (tid == 0) sK = Ks[which * BB + b];

  // ---- TDM async tensor load: each of the 8 waves DMAs its own 2x4096 f32
  // subtile into LDS. pad_interval=7 / pad_amount=0 inserts 1 pad dword every
  // 256 dwords, so padded position of logical element i is i + (i>>8).
  {
    const int wave = tid >> 5;             // wave-uniform
    uint64_t gaddr = (uint64_t)(uintptr_t)
        (U_base + (size_t)which * BB * NN + (size_t)b * NN + (size_t)wave * 8192);
    uint32_t ldsoff = __builtin_amdgcn_groupstaticsize()   // dynamic-LDS start
                    + (uint32_t)wave * 8224u * 4u;         // 8192 + 32 pad dwords
    v4u g0;
    g0[0] = 1u;                                      // count=1, is_restore=0
    g0[1] = ldsoff;                                  // lds_addr (bytes)
    g0[2] = (uint32_t)gaddr;                         // global_addr[31:0]
    g0[3] = (uint32_t)(gaddr >> 32) | 0x80000000u;   // global_addr[56:32] | type=2
    uint64_t q0 = (2ull << 16)                       // data_size = 4 bytes
                | (1ull << 20)                       // pad_enable
                | (7ull << 22)                       // pad_interval: 256 dwords
                | (0ull << 25)                       // pad_amount: 1 dword
                | (4096ull << 48);                   // tensor_dim0 lo16 = 4096
    uint64_t q1 = (0ull)                             // tensor_dim0 hi16
                | (2ull << 16)                       // tensor_dim1 = 2 rows
                | (4096ull << 48);                   // tile_dim0 = 4096
    uint64_t q2 = 2ull                               // tile_dim1 = 2
                | (4096ull << 32);                   // tensor_dim0_stride lo32
    v8i g1;
    g1[0] = (int)(uint32_t)q0; g1[1] = (int)(uint32_t)(q0 >> 32);
    g1[2] = (int)(uint32_t)q1; g1[3] = (int)(uint32_t)(q1 >> 32);
    g1[4] = (int)(uint32_t)q2; g1[5] = (int)(uint32_t)(q2 >> 32);
    g1[6] = 0;                 g1[7] = 0;            // dim1_stride unused (2D)
    v4i z4 = {0, 0, 0, 0};                           // groups 2/3 unused (2D)
    v8i z8 = {0, 0, 0, 0, 0, 0, 0, 0};               // extra group (clang-23 form)
    __builtin_amdgcn_tensor_load_to_lds(g0, g1, z4, z4, z8, 0);
  }

  // ---- Overlap the in-flight DMA with the dR broadcast store.
  if (which == 0) {
    float v = dRb[b];
    float4 vv = make_float4(v, v, v, v);
    float4* o4 = (float4*)(out + (size_t)2 * BB * NN + (size_t)b * NN);
    for (int i = tid; i < NN / 4; i += 256) o4[i] = vv;
  }

  __builtin_amdgcn_s_wait_tensorcnt(0);    // each wave waits on its own DMA
  __syncthreads();

  // ---- F = log(clamp(u)) + logT[t]; convert in place to a total-order u32
  // key (larger float -> larger key). Strided access: padded addr = tid+257*s,
  // consecutive lanes -> consecutive banks.
  for (int i = tid; i < NN; i += 256) {
    int p = i + (i >> 8);                           // padded LDS position
    float u = clampf01(__uint_as_float(keys[p]));
    float f = logf(u) + logT[i >> 12];              // t = i / 4096
    uint32_t bits = __float_as_uint(f);
    keys[p] = (bits & 0x80000000u) ? ~bits : (bits | 0x80000000u);
  }
  __syncthreads();

  const int K = sK;
  float* orow = out + (size_t)which * BB * NN + (size_t)b * NN;

  if (K <= 0 || K >= NN) {                 // trivial masks
    float val = (K >= NN) ? 1.0f : 0.0f;
    float4 vv = make_float4(val, val, val, val);
    float4* o4 = (float4*)orow;
    for (int i = tid; i < NN / 4; i += 256) o4[i] = vv;
    return;
  }

  // ---- Exact 4-pass MSB->LSB radix select for the K-th largest key,
  // entirely from LDS (row never re-read from HBM).
  if (tid == 0) sRem = (uint32_t)K;
  uint32_t prefix = 0;
  const int      shifts[4] = {24, 16, 8, 0};
  const uint32_t hiMask[4] = {0u, 0xFF000000u, 0xFFFF0000u, 0xFFFFFF00u};
#pragma unroll 1
  for (int pass = 0; pass < 4; ++pass) {
    hist[tid] = 0;
    __syncthreads();
    const int sh = shifts[pass];
    const uint32_t hm = hiMask[pass];
    for (int i = tid; i < NN; i += 256) {
      uint32_t k = keys[i + (i >> 8)];
      if (((k ^ prefix) & hm) == 0)
        atomicAdd(&hist[(k >> sh) & 0xFFu], 1u);
    }
    __syncthreads();
    if (tid == 0) {
      uint32_t rem = sRem;
      int dsel = 0;
      for (int d = 255; d >= 0; --d) {     // scan from the top digit down
        uint32_t c = hist[d];
        if (rem <= c) { dsel = d; break; }
        rem -= c;
      }
      sDigit = (uint32_t)dsel;
      sRem   = rem;
    }
    __syncthreads();
    prefix |= (sDigit << sh);
  }
  const uint32_t thr  = prefix;            // exact K-th largest key
  const uint32_t need = sRem;              // #keys == thr to keep (lowest index first)

  // ---- Stable tie resolution (matches jnp stable argsort: equal scores are
  // ranked by ascending index) + mask emit. Thread owns a 256-element chunk;
  // padded chunk base tid*257 makes these reads bank-conflict free.
  uint32_t eqc = 0;
  for (int j = 0; j < 256; ++j) eqc += (keys[tid * 257 + j] == thr) ? 1u : 0u;
  eqcnt[tid] = eqc;
  __syncthreads();
  uint32_t rank = 0;
  for (int j = 0; j < tid; ++j) rank += eqcnt[j];   // exclusive prefix of equals
  float4* o4 = (float4*)(orow + (tid << 8));
  for (int j = 0; j < 256; j += 4) {
    float r[4];
#pragma unroll
    for (int q = 0; q < 4; ++q) {
      uint32_t k = keys[tid * 257 + j + q];
      bool sel = (k > thr) || ((k == thr) && (rank < need));
      if (k == thr) ++rank;
      r[q] = sel ? 1.0f : 0.0f;
    }
    o4[j >> 2] = make_float4(r[0], r[1], r[2], r[3]);
  }
}

// --------------------------------------------------------------------------
extern "C" void kernel_launch(void* const* d_in, const int* in_sizes, int n_in,
                              void* d_out, int out_size, void* d_ws, size_t ws_size,
                              hipStream_t stream) {
  (void)in_sizes; (void)n_in; (void)out_size; (void)ws_size;
  // d_in[0] = B (scalar, ignored; configuration is fixed)
  const float* U_base    = (const float*)d_in[1];   // [2,B,N] f32
  const float* U_event_t = (const float*)d_in[2];   // [B,T]   f32
  const float* U_rate    = (const float*)d_in[3];   // [2,B]   f32
  float* out = (float*)d_out;                       // [src|tgt|dR] 3*B*N f32

  float* logU   = (float*)d_ws;            // B*T
  float* log1mU = logU + BB * TT;          // B*T
  float* dRb    = log1mU + BB * TT;        // B
  int*   Ks     = (int*)(dRb + BB);        // 2*B

  prep_kernel<<<1, BB, 0, stream>>>(U_event_t, U_rate, logU, log1mU, dRb, Ks);

  // 263,168 B dynamic LDS (row + 1 pad dword per 256) <= 320 KB WGP LDS
  const size_t ldsBytes = (size_t)(NN + (NN >> 8)) * sizeof(uint32_t);
  topk_mask_kernel<<<2 * BB, 256, ldsBytes, stream>>>(U_base, logU, log1mU,
                                                      dRb, Ks, out);
}